// CausalConv1d_59760174956725
// MI455X (gfx1250) — compile-verified
//
#include <hip/hip_runtime.h>
#include <hip/hip_bf16.h>

typedef __attribute__((ext_vector_type(4))) float vfloat4;

#define TPB      256   // threads per block (8 waves, wave32)
#define S_STAGE  8     // timesteps per async stage
#define N_STAGE  8     // stages per chunk
#define TT       (S_STAGE * N_STAGE)   // 64 timesteps per thread-chunk

// ---------------------------------------------------------------------------
// Main streaming kernel: depthwise causal conv, K=4, async global->LDS staging
// Each thread owns 4 channels (one float4 column) and TT consecutive steps.
// ---------------------------------------------------------------------------
__global__ __launch_bounds__(TPB) void conv_async_kernel(
    const float* __restrict__ x, const float* __restrict__ cs,
    const float* __restrict__ w, const float* __restrict__ bias,
    float* __restrict__ y, int T, int C) {
  __shared__ vfloat4 tile[2][S_STAGE][TPB];   // 64 KB double buffer

  const int tid = threadIdx.x;
  const int CG  = C >> 2;                     // float4 groups per row
  const int cg  = blockIdx.x * TPB + tid;     // this thread's channel group
  const int t0  = blockIdx.y * TT;            // first timestep of chunk
  const int b   = blockIdx.z;
  const int c0  = cg << 2;

  // Load weights (C,K=4) and transpose to tap-major vectors over 4 channels.
  const float* wp = w + (size_t)c0 * 4;
  vfloat4 wr0 = *(const vfloat4*)(wp + 0);
  vfloat4 wr1 = *(const vfloat4*)(wp + 4);
  vfloat4 wr2 = *(const vfloat4*)(wp + 8);
  vfloat4 wr3 = *(const vfloat4*)(wp + 12);
  vfloat4 w0 = {wr0.x, wr1.x, wr2.x, wr3.x};
  vfloat4 w1 = {wr0.y, wr1.y, wr2.y, wr3.y};
  vfloat4 w2 = {wr0.z, wr1.z, wr2.z, wr3.z};
  vfloat4 w3 = {wr0.w, wr1.w, wr2.w, wr3.w};
  vfloat4 bv = *(const vfloat4*)(bias + c0);

  const vfloat4* xg = (const vfloat4*)x + ((size_t)b * T) * CG + cg;
  vfloat4*       yg = (vfloat4*)y + ((size_t)b * T + t0) * CG + cg;

  // Rolling window: h0,h1,h2 = full[t0], full[t0+1], full[t0+2]
  vfloat4 h0, h1, h2;
  if (t0 == 0) {
    // left context = conv_state[b, c, 1..3]   (cs layout (B,C,K), K==4)
    const float* p = cs + ((size_t)b * C + c0) * 4;
    h0 = vfloat4{p[1], p[5], p[9],  p[13]};
    h1 = vfloat4{p[2], p[6], p[10], p[14]};
    h2 = vfloat4{p[3], p[7], p[11], p[15]};
  } else {
    h0 = xg[(size_t)(t0 - 3) * CG];
    h1 = xg[(size_t)(t0 - 2) * CG];
    h2 = xg[(size_t)(t0 - 1) * CG];
  }

  const unsigned ldsBase  = (unsigned)(uintptr_t)&tile[0][0][0];
  const unsigned bufBytes = S_STAGE * TPB * 16u;
  const unsigned laneOff  = (unsigned)tid * 16u;

  // Prefetch stage 0 into buffer 0 (async DMA global -> LDS, per-lane B128).
  {
    const vfloat4* gp = xg + (size_t)t0 * CG;
#pragma unroll
    for (int j = 0; j < S_STAGE; ++j) {
      unsigned loff = ldsBase + (unsigned)j * (TPB * 16u) + laneOff;
      const vfloat4* g = gp + (size_t)j * CG;
      asm volatile("global_load_async_to_lds_b128 %0, %1, off"
                   :: "v"(loff), "v"(g) : "memory");
    }
  }

  for (int s = 0; s < N_STAGE; ++s) {
    if (s + 1 < N_STAGE) {
      // Issue next stage into the other buffer, then wait for current stage.
      const vfloat4* gp = xg + (size_t)(t0 + (s + 1) * S_STAGE) * CG;
      unsigned nb = ((unsigned)(s + 1) & 1u) * bufBytes;
#pragma unroll
      for (int j = 0; j < S_STAGE; ++j) {
        unsigned loff = ldsBase + nb + (unsigned)j * (TPB * 16u) + laneOff;
        const vfloat4* g = gp + (size_t)j * CG;
        asm volatile("global_load_async_to_lds_b128 %0, %1, off"
                     :: "v"(loff), "v"(g) : "memory");
      }
      // 8 newly issued outstanding; <=8 means the current stage completed
      // (async loads complete in order).
      asm volatile("s_wait_asynccnt %0" :: "n"(S_STAGE) : "memory");
    } else {
      asm volatile("s_wait_asynccnt 0" ::: "memory");
    }

    const int cb = s & 1;
#pragma unroll
    for (int j = 0; j < S_STAGE; ++j) {
      vfloat4 xv  = tile[cb][j][tid];     // ds_load_b128, own slot -> no barrier
      vfloat4 acc = bv;
      acc += h0 * w0;
      acc += h1 * w1;
      acc += h2 * w2;
      acc += xv * w3;
      __builtin_nontemporal_store(acc, yg);   // global_store_b128 th:NT (y is write-once)
      yg += CG;
      h0 = h1; h1 = h2; h2 = xv;
    }
  }
}

// ---------------------------------------------------------------------------
// Generic fallback (any B,T,C,K / divisibility) — correctness safety net.
// ---------------------------------------------------------------------------
__global__ void conv_generic(const float* __restrict__ x, const float* __restrict__ cs,
                             const float* __restrict__ w, const float* __restrict__ bias,
                             float* __restrict__ y, int B, int T, int C, int K) {
  size_t i = (size_t)blockIdx.x * blockDim.x + threadIdx.x;
  size_t total = (size_t)B * T * C;
  if (i >= total) return;
  int c = (int)(i % C);
  size_t bt = i / C;
  int t = (int)(bt % T);
  int b = (int)(bt / T);
  float acc = bias[c];
  for (int k = 0; k < K; ++k) {
    int tf = t + k - (K - 1);
    float v = (tf >= 0) ? x[((size_t)b * T + tf) * C + c]
                        : cs[((size_t)b * C + c) * K + (K + tf)];
    acc += v * w[c * K + k];
  }
  y[i] = acc;
}

// ---------------------------------------------------------------------------
// Final rolling state: last K columns of cat(conv_state, x^T)  -> (B,C,K)
// ---------------------------------------------------------------------------
__global__ void state_kernel(const float* __restrict__ x, const float* __restrict__ cs,
                             float* __restrict__ st, int B, int T, int C, int K) {
  int i = blockIdx.x * blockDim.x + threadIdx.x;   // b*C + c
  if (i >= B * C) return;
  int b = i / C, c = i % C;
  for (int k = 0; k < K; ++k) {
    int src = T - K + k;   // index into x along T
    float v = (src >= 0) ? x[((size_t)b * T + src) * C + c]
                         : cs[((size_t)b * C + c) * K + (K + src)];
    st[(size_t)i * K + k] = v;
  }
}

extern "C" void kernel_launch(void* const* d_in, const int* in_sizes, int n_in,
                              void* d_out, int out_size, void* d_ws, size_t ws_size,
                              hipStream_t stream) {
  const float* x    = (const float*)d_in[0];
  const float* cs   = (const float*)d_in[1];
  const float* w    = (const float*)d_in[2];
  const float* bias = (const float*)d_in[3];

  // Recover shapes from flat sizes: bias=(C,), weight=(C,K), cs=(B,C,K), x=(B,T,C)
  int C = in_sizes[3];
  int K = in_sizes[2] / C;
  int B = in_sizes[1] / (C * K);
  int T = in_sizes[0] / (B * C);

  float* y  = (float*)d_out;
  float* st = y + (size_t)B * T * C;

  if (K == 4 && (C % (4 * TPB)) == 0 && (T % TT) == 0 && T >= 4) {
    dim3 grid(C / (4 * TPB), T / TT, B);
    conv_async_kernel<<<grid, dim3(TPB), 0, stream>>>(x, cs, w, bias, y, T, C);
  } else {
    size_t total = (size_t)B * T * C;
    int blocks = (int)((total + 255) / 256);
    conv_generic<<<blocks, 256, 0, stream>>>(x, cs, w, bias, y, B, T, C, K);
  }

  int n = B * C;
  state_kernel<<<(n + 255) / 256, 256, 0, stream>>>(x, cs, st, B, T, C, K);
}